// DistLoss_69054484185810
// MI455X (gfx1250) — compile-verified
//
#include <hip/hip_runtime.h>

// CDNA5 / gfx1250. wave32 only.
typedef float v2f __attribute__((ext_vector_type(2)));
typedef float v8f __attribute__((ext_vector_type(8)));

#define WG_THREADS   256
#define WAVES_PER_WG 8
#define TILE         16

// Each wave: one 16-target column tile; sweeps all surface points in 16-row
// tiles via V_WMMA_F32_16X16X4_F32, computing D = ||s||^2 - 2 s.t per tile
// (A is pre-scaled by -2 in LDS, C carries ||s||^2), then running min.
__global__ __launch_bounds__(WG_THREADS)
void dist_min_kernel(const float* __restrict__ surf,   // [nSurf][3]
                     const float* __restrict__ tgt,    // [nTgt][3]
                     float* __restrict__ partial,      // [gridDim.x]
                     int nSurf)
{
    extern __shared__ float smem[];  // nSurf * 4 floats: {-2x, -2y, -2z, ||s||^2}

    const int tid = threadIdx.x;

    // ---- Stage all surface points into LDS (pre-scaled, with a2 in .w) ----
    for (int i = tid; i < nSurf; i += WG_THREADS) {
        float x = surf[3 * i + 0];
        float y = surf[3 * i + 1];
        float z = surf[3 * i + 2];
        v2f lo = { -2.0f * x, -2.0f * y };
        v2f hi = { -2.0f * z, x * x + y * y + z * z };
        *(v2f*)(smem + 4 * i)     = lo;   // 8B aligned
        *(v2f*)(smem + 4 * i + 2) = hi;
    }
    __syncthreads();

    const int lane = tid & 31;      // wave32
    const int wave = tid >> 5;
    const int half = lane >> 4;     // 0 -> K rows 0..1 / M rows 0..7 ; 1 -> K 2..3 / M 8..15
    const int l16  = lane & 15;

    const int col0 = (blockIdx.x * WAVES_PER_WG + wave) * TILE;
    const int t    = col0 + l16;    // this lane's target column

    // ---- B matrix (4x16, loop-invariant). ISA layout:
    //   lanes 0..15 : V0=B[0][n]=t.x, V1=B[1][n]=t.y
    //   lanes 16..31: V0=B[2][n]=t.z, V1=B[3][n]=0  (zero-padded K=3 row)
    const float bx = tgt[3 * t + 0];
    const float by = tgt[3 * t + 1];
    const float bz = tgt[3 * t + 2];
    v2f b;
    b.x = half ? bz : bx;
    b.y = half ? 0.0f : by;
    const float b2part = b.x * b.x + b.y * b.y;  // lower: x^2+y^2, upper: z^2

    const float INF = 3.402823466e+38f;
    float mv[8];
#pragma unroll
    for (int v = 0; v < 8; ++v) mv[v] = INF;

    // ---- Main sweep: 16 surface rows per iteration ----
    for (int p0 = 0; p0 < nSurf; p0 += TILE) {
        // A tile: lane row = l16; lower half loads {-2x,-2y} (K=0,1),
        // upper half loads {-2z, a2} (K=2,3; K=3 column hits B's zero row).
        const float* ap = smem + 4 * (p0 + l16) + 2 * half;
        v2f a = *(const v2f*)ap;

        // C tile: C[m][n] = ||s_m||^2 ; lane needs rows m = 8*half + v.
        // Same address across each lane-half -> LDS broadcast reads.
        const float* a2p = smem + 4 * (p0 + 8 * half) + 3;
        v8f c;
#pragma unroll
        for (int v = 0; v < 8; ++v) c[v] = a2p[4 * v];

        // D = A*B + C = ||s||^2 - 2 s.t  for a 16x16 tile
        v8f d = __builtin_amdgcn_wmma_f32_16x16x4_f32(
            /*neg_a=*/false, a, /*neg_b=*/false, b,
            /*c_mod=*/(short)0, c, /*reuse_a=*/false, /*reuse_b=*/false);

#pragma unroll
        for (int v = 0; v < 8; ++v) mv[v] = fminf(mv[v], d[v]);
    }

    // ---- Reduce: min over this lane's 8 row-slots ----
    float m8 = mv[0];
#pragma unroll
    for (int v = 1; v < 8; ++v) m8 = fminf(m8, mv[v]);

    // Combine the two lane-halves (rows 0-7 mod 16 vs 8-15 mod 16)
    float colmin = fminf(m8, __shfl_xor(m8, 16, 32));

    // ||t||^2 for this column (x^2+y^2 from lower half, z^2 from upper)
    float b2 = b2part + __shfl_xor(b2part, 16, 32);

    float val = colmin + b2;        // nearest squared distance for target t
    if (half) val = 0.0f;           // column duplicated across halves: count once

    // Wave sum over 16 columns (upper half contributes 0)
#pragma unroll
    for (int m = 16; m >= 1; m >>= 1) val += __shfl_xor(val, m, 32);

    // ---- Per-workgroup partial (reuse LDS after barrier) ----
    __syncthreads();
    if (lane == 0) smem[wave] = val;
    __syncthreads();
    if (tid == 0) {
        float s = 0.0f;
#pragma unroll
        for (int w = 0; w < WAVES_PER_WG; ++w) s += smem[w];
        partial[blockIdx.x] = s;
    }
}

__global__ void final_sum_kernel(const float* __restrict__ partial,
                                 float* __restrict__ out, int n)
{
    if (threadIdx.x == 0 && blockIdx.x == 0) {
        float s = 0.0f;
        for (int i = 0; i < n; ++i) s += partial[i];
        out[0] = s;
    }
}

extern "C" void kernel_launch(void* const* d_in, const int* in_sizes, int n_in,
                              void* d_out, int out_size, void* d_ws, size_t ws_size,
                              hipStream_t stream) {
    const float* surf = (const float*)d_in[0];   // [S*N, 3] = [16384, 3]
    const float* tgt  = (const float*)d_in[1];   // [M, 3]   = [16384, 3]
    const int nSurf = in_sizes[0] / 3;           // 16384 surface points
    const int nTgt  = in_sizes[1] / 3;           // 16384 targets

    float* partial = (float*)d_ws;

    const int nWG = nTgt / (TILE * WAVES_PER_WG);            // 128 workgroups
    const size_t ldsBytes = (size_t)nSurf * 4 * sizeof(float); // 256 KB / WGP (<= 320 KB)

    // 256 KB dynamic LDS exceeds the default cap; raise it (not a stream op,
    // safe under graph capture; deterministic every call).
    hipFuncSetAttribute((const void*)dist_min_kernel,
                        hipFuncAttributeMaxDynamicSharedMemorySize,
                        (int)ldsBytes);

    dist_min_kernel<<<nWG, WG_THREADS, ldsBytes, stream>>>(surf, tgt, partial, nSurf);
    final_sum_kernel<<<1, 32, 0, stream>>>(partial, (float*)d_out, nWG);
}